// MultiTaskQuestionGeneratorForDecoder_42468636623043
// MI455X (gfx1250) — compile-verified
//
#include <hip/hip_runtime.h>
#include <hip/hip_bf16.h>
#include <stdint.h>
#include <math.h>

typedef __attribute__((ext_vector_type(2))) float v2f;
typedef __attribute__((ext_vector_type(8))) float v8f;

#define HOPS  3
#define VOCAB 100000
#define E     128
#define H     128
#define B     16
#define S     2048
#define M     4
#define NE    4000
#define NR    200

// ---------------------------------------------------------------- RNG (deterministic hash-based gumbel)
__device__ __forceinline__ float urand01(uint32_t salt, uint32_t b, uint32_t i) {
    uint32_t x = (salt * 0x9E3779B9u) ^ (b * 0x85EBCA6Bu) ^ (i * 0xC2B2AE35u) ^ 0x6C8E9CF5u;
    x ^= x >> 16; x *= 0x7FEB352Du;
    x ^= x >> 15; x *= 0x846CA68Bu;
    x ^= x >> 16;
    float r = (float)(x >> 8) * (1.0f / 16777216.0f);   // [0, 1)
    return r < 1e-10f ? 1e-10f : r;
}
__device__ __forceinline__ float gumbel_noise(uint32_t salt, uint32_t b, uint32_t i) {
    float u = urand01(salt, b, i);
    return -logf(-logf(u));
}

// ---------------------------------------------------------------- h = hidden @ Wm^T + Wb  (WMMA f32 16x16x4)
// One block, 8 waves; wave j computes output columns [16j, 16j+16). Also writes u = h.
__global__ void __launch_bounds__(256) k_hidden_proj(
    const float* __restrict__ hidden, const float* __restrict__ Wm,
    const float* __restrict__ Wb, float* __restrict__ h, float* __restrict__ u)
{
    const int lane = threadIdx.x & 31;
    const int wave = threadIdx.x >> 5;      // 0..7 -> N tile
    const int row  = lane & 15;
    const int kb   = (lane >> 4) << 1;      // 0 or 2
    const int n0   = wave * 16;

    v8f c = {0.f, 0.f, 0.f, 0.f, 0.f, 0.f, 0.f, 0.f};
#pragma unroll
    for (int kk = 0; kk < H; kk += 4) {
        v2f a, b;
        a[0] = hidden[row * H + kk + kb + 0];       // A[m=row][k]
        a[1] = hidden[row * H + kk + kb + 1];
        b[0] = Wm[(n0 + row) * H + kk + kb + 0];    // B[k][n] = Wm[n][k]
        b[1] = Wm[(n0 + row) * H + kk + kb + 1];
        c = __builtin_amdgcn_wmma_f32_16x16x4_f32(false, a, false, b, (short)0, c, false, false);
    }
    const int colBase = n0 + (lane & 15);
    const int rowBase = (lane >> 4) * 8;
#pragma unroll
    for (int r = 0; r < 8; ++r) {
        const int m = rowBase + r;
        const float val = c[r] + Wb[colBase];
        h[m * H + colBase] = val;
        u[m * H + colBase] = val;
    }
}

// ---------------------------------------------------------------- qh = h @ W3w^T + W3b  (16 x 4000, WMMA)
__global__ void __launch_bounds__(256) k_qh_proj(
    const float* __restrict__ h, const float* __restrict__ W3w,
    const float* __restrict__ W3b, float* __restrict__ out)
{
    const int lane = threadIdx.x & 31;
    const int wave = threadIdx.x >> 5;
    const int tile = blockIdx.x * 8 + wave;         // wave-uniform: EXEC stays full
    if (tile >= NE / 16) return;
    const int n0  = tile * 16;
    const int row = lane & 15;
    const int kb  = (lane >> 4) << 1;

    v8f c = {0.f, 0.f, 0.f, 0.f, 0.f, 0.f, 0.f, 0.f};
#pragma unroll
    for (int kk = 0; kk < H; kk += 4) {
        v2f a, b;
        a[0] = h[row * H + kk + kb + 0];
        a[1] = h[row * H + kk + kb + 1];
        b[0] = W3w[(size_t)(n0 + row) * H + kk + kb + 0];
        b[1] = W3w[(size_t)(n0 + row) * H + kk + kb + 1];
        c = __builtin_amdgcn_wmma_f32_16x16x4_f32(false, a, false, b, (short)0, c, false, false);
    }
    const int colBase = n0 + (lane & 15);
    const int rowBase = (lane >> 4) * 8;
#pragma unroll
    for (int r = 0; r < 8; ++r) {
        const int m = rowBase + r;
        out[m * NE + colBase] = c[r] + W3b[colBase];
    }
}

// ---------------------------------------------------------------- sp (16x2) and qt (16x200): tiny VALU GEMVs
__global__ void __launch_bounds__(256) k_small_proj(
    const float* __restrict__ h,
    const float* __restrict__ W1w, const float* __restrict__ W1b,
    const float* __restrict__ W4w, const float* __restrict__ W4b,
    float* __restrict__ sp, float* __restrict__ qt)
{
    const int t = blockIdx.x * 256 + threadIdx.x;
    if (t < B * NR) {
        const int b = t / NR, r = t % NR;
        float acc = W4b[r];
        for (int k = 0; k < H; ++k) acc += h[b * H + k] * W4w[r * H + k];
        qt[b * NR + r] = acc;
    }
    if (t < B * 2) {
        const int b = t >> 1, r = t & 1;
        float acc = W1b[r];
        for (int k = 0; k < H; ++k) acc += h[b * H + k] * W1w[r * H + k];
        sp[b * 2 + r] = acc;
    }
}

// ---------------------------------------------------------------- gumbel-softmax hard (straight-through fwd)
// One block per batch row. out = hard - y + y with y = softmax(logits + g).
__global__ void __launch_bounds__(256) k_gumbel_hard(
    const float* __restrict__ logits, float* __restrict__ out, int N, uint32_t salt)
{
    __shared__ float smax[256];
    __shared__ int   sidx[256];
    const int b = blockIdx.x, t = threadIdx.x;
    const int base = b * N;

    float bm = -INFINITY; int bi = 0x7fffffff;
    for (int i = t; i < N; i += 256) {
        const float z = logits[base + i] + gumbel_noise(salt, (uint32_t)b, (uint32_t)i);
        if (z > bm || (z == bm && i < bi)) { bm = z; bi = i; }
    }
    smax[t] = bm; sidx[t] = bi; __syncthreads();
    for (int o = 128; o; o >>= 1) {
        if (t < o) {
            if (smax[t + o] > smax[t] || (smax[t + o] == smax[t] && sidx[t + o] < sidx[t])) {
                smax[t] = smax[t + o]; sidx[t] = sidx[t + o];
            }
        }
        __syncthreads();
    }
    const float zmax = smax[0];
    const int   amax = sidx[0];
    __syncthreads();

    float sm = 0.f;
    for (int i = t; i < N; i += 256) {
        const float z = logits[base + i] + gumbel_noise(salt, (uint32_t)b, (uint32_t)i);
        sm += expf(z - zmax);
    }
    smax[t] = sm; __syncthreads();
    for (int o = 128; o; o >>= 1) { if (t < o) smax[t] += smax[t + o]; __syncthreads(); }
    const float inv = 1.0f / smax[0];

    for (int i = t; i < N; i += 256) {
        const float z = logits[base + i] + gumbel_noise(salt, (uint32_t)b, (uint32_t)i);
        const float y = expf(z - zmax) * inv;
        const float hard = (i == amax) ? 1.0f : 0.0f;
        out[base + i] = (hard - y) + y;
    }
}

// ---------------------------------------------------------------- fused gather + dot:  logit[b,s] = <sum_m C[story], u[b]>
// One wave per (b,s); each lane owns a float4 chunk of E=128.
__global__ void __launch_bounds__(256) k_hop_logits(
    const int* __restrict__ story, const float* __restrict__ Ctab,
    const float* __restrict__ u, float* __restrict__ logit)
{
    const int w    = blockIdx.x * 8 + (threadIdx.x >> 5);   // 0..32767
    const int lane = threadIdx.x & 31;
    const int b = w >> 11;          // / 2048
    const int s = w & (S - 1);

    const int* ids = story + (size_t)(b * S + s) * M;
    float4 acc = make_float4(0.f, 0.f, 0.f, 0.f);
#pragma unroll
    for (int m = 0; m < M; ++m) {
        const int id = ids[m];
        const float4 v = ((const float4*)(Ctab + (size_t)id * E))[lane];
        acc.x += v.x; acc.y += v.y; acc.z += v.z; acc.w += v.w;
    }
    const float4 uu = ((const float4*)(u + b * E))[lane];
    float p = acc.x * uu.x + acc.y * uu.y + acc.z * uu.z + acc.w * uu.w;
#pragma unroll
    for (int off = 16; off; off >>= 1) p += __shfl_xor(p, off, 32);
    if (lane == 0) logit[b * S + s] = p;
}

// ---------------------------------------------------------------- softmax over S per batch row
__global__ void __launch_bounds__(256) k_softmax_S(
    const float* __restrict__ logit, float* __restrict__ prob)
{
    __shared__ float red[256];
    const int b = blockIdx.x, t = threadIdx.x;
    float mx = -INFINITY;
    for (int s = t; s < S; s += 256) mx = fmaxf(mx, logit[b * S + s]);
    red[t] = mx; __syncthreads();
    for (int o = 128; o; o >>= 1) { if (t < o) red[t] = fmaxf(red[t], red[t + o]); __syncthreads(); }
    mx = red[0]; __syncthreads();
    float sm = 0.f;
    for (int s = t; s < S; s += 256) sm += expf(logit[b * S + s] - mx);
    red[t] = sm; __syncthreads();
    for (int o = 128; o; o >>= 1) { if (t < o) red[t] += red[t + o]; __syncthreads(); }
    const float inv = 1.0f / red[0];
    for (int s = t; s < S; s += 256) prob[b * S + s] = expf(logit[b * S + s] - mx) * inv;
}

// ---------------------------------------------------------------- fused gather + weighted sum (stage 1: partials)
// Block = (b, chunk of 32 s-values); thread = embedding index e.
__global__ void __launch_bounds__(128) k_update_partial(
    const int* __restrict__ story, const float* __restrict__ Ctab,
    const float* __restrict__ prob, float* __restrict__ partial)
{
    const int blk = blockIdx.x;         // b*64 + chunk
    const int b = blk >> 6, chunk = blk & 63;
    const int e = threadIdx.x;
    const int s0 = chunk * 32;
    float acc = 0.f;
    for (int i = 0; i < 32; ++i) {
        const int s = s0 + i;
        const float p = prob[b * S + s];
        const int* ids = story + (size_t)(b * S + s) * M;
        float sum = 0.f;
#pragma unroll
        for (int m = 0; m < M; ++m) sum += Ctab[(size_t)ids[m] * E + e];
        acc += p * sum;
    }
    partial[(size_t)blk * E + e] = acc;
}

// stage 2: u[b][e] += sum over 64 chunks
__global__ void __launch_bounds__(128) k_reduce_u(
    const float* __restrict__ partial, float* __restrict__ u)
{
    const int b = blockIdx.x, e = threadIdx.x;
    float acc = 0.f;
    for (int c = 0; c < 64; ++c) acc += partial[(size_t)((b << 6) + c) * E + e];
    u[b * E + e] += acc;
}

// ---------------------------------------------------------------- masked sigmoid of final-hop logits
__global__ void __launch_bounds__(256) k_mask_sig(
    const float* __restrict__ logit, const int* __restrict__ lengths, float* __restrict__ out)
{
    const int t = blockIdx.x * 256 + threadIdx.x;
    if (t >= B * S) return;
    const int b = t / S, s = t % S;
    const float v = 1.0f / (1.0f + expf(-logit[t]));
    out[t] = (s < lengths[b]) ? v : 0.0f;
}

// ---------------------------------------------------------------- launcher
extern "C" void kernel_launch(void* const* d_in, const int* in_sizes, int n_in,
                              void* d_out, int out_size, void* d_ws, size_t ws_size,
                              hipStream_t stream) {
    (void)in_sizes; (void)n_in; (void)out_size; (void)ws_size;

    const int*   story   = (const int*)  d_in[0];
    const int*   lengths = (const int*)  d_in[1];
    const float* hidden  = (const float*)d_in[2];
    // d_in[3] = global_pointer (unused by reference)
    const float* C       = (const float*)d_in[4];
    const float* Wm      = (const float*)d_in[5];
    const float* Wb      = (const float*)d_in[6];
    const float* W1w     = (const float*)d_in[7];
    const float* W1b     = (const float*)d_in[8];
    const float* W3w     = (const float*)d_in[9];
    const float* W3b     = (const float*)d_in[10];
    const float* W4w     = (const float*)d_in[11];
    const float* W4b     = (const float*)d_in[12];

    float* out     = (float*)d_out;
    float* sp      = out + 0;        // [16,2]
    float* sp_act  = out + 32;       // [16,2]
    float* qh      = out + 64;       // [16,4000]
    float* qh_act  = out + 64064;    // [16,4000]
    float* qt      = out + 128064;   // [16,200]
    float* qt_act  = out + 131264;   // [16,200]
    float* masked  = out + 134464;   // [16,2048]

    float* ws = (float*)d_ws;
    float* h  = ws;                  // 2048
    float* u  = ws + 2048;           // 2048
    float* lg = ws + 4096;           // 32768
    float* pr = ws + 36864;          // 32768
    float* pt = ws + 69632;          // 131072  (total < 1 MB)

    const size_t TBL = (size_t)VOCAB * E;

    k_hidden_proj<<<1, 256, 0, stream>>>(hidden, Wm, Wb, h, u);
    k_qh_proj<<<32, 256, 0, stream>>>(h, W3w, W3b, qh);
    k_small_proj<<<13, 256, 0, stream>>>(h, W1w, W1b, W4w, W4b, sp, qt);
    k_gumbel_hard<<<16, 256, 0, stream>>>(sp, sp_act, 2,  1u);
    k_gumbel_hard<<<16, 256, 0, stream>>>(qh, qh_act, NE, 2u);
    k_gumbel_hard<<<16, 256, 0, stream>>>(qt, qt_act, NR, 3u);

    // hops 0 and 1: logits -> softmax -> u update (hop-2 update & softmax are dead code)
    for (int hop = 0; hop < 2; ++hop) {
        k_hop_logits<<<4096, 256, 0, stream>>>(story, C + (size_t)hop * TBL, u, lg);
        k_softmax_S<<<16, 256, 0, stream>>>(lg, pr);
        k_update_partial<<<1024, 128, 0, stream>>>(story, C + (size_t)(hop + 1) * TBL, pr, pt);
        k_reduce_u<<<16, 128, 0, stream>>>(pt, u);
    }
    // hop 2: logits with C[2], then sigmoid + length mask
    k_hop_logits<<<4096, 256, 0, stream>>>(story, C + 2 * TBL, u, lg);
    k_mask_sig<<<128, 256, 0, stream>>>(lg, lengths, masked);
}